// _PCL_Losses_43550968381611
// MI455X (gfx1250) — compile-verified
//
#include <hip/hip_runtime.h>
#include <hip/hip_bf16.h>

#define CCH 21  // channels

typedef __attribute__((ext_vector_type(2))) float v2f;
typedef __attribute__((ext_vector_type(8))) float v8f;

// Full-wave32 sum using V_WMMA_F32_16X16X4_F32 as the reduction engine.
// A = ones(16x4), B vgpr0 = per-lane partial, B vgpr1 = 0.
// D[m][n] = sum_k B[k][n]; the 16 columns partition all 32 partials, so
// summing row M=0 (held in lanes 0..15 of D vgpr0, mirrored in 16..31)
// across lanes yields the wave total in every lane.
// EXEC must be all ones: call uniformly from every thread of a full block.
__device__ __forceinline__ float wave_reduce_sum_wmma(float x) {
  v2f a; a[0] = 1.0f; a[1] = 1.0f;   // A-matrix: all ones
  v2f b; b[0] = x;    b[1] = 0.0f;   // B-matrix: partials in first VGPR
  v8f c = {};
  c = __builtin_amdgcn_wmma_f32_16x16x4_f32(
      /*neg_a=*/false, a, /*neg_b=*/false, b,
      /*c_mod=*/(short)0, c, /*reuse_a=*/false, /*reuse_b=*/false);
  float q = c[0];                    // lane l holds column-sum q[l & 15]
  q += __shfl_xor(q, 1);
  q += __shfl_xor(q, 2);
  q += __shfl_xor(q, 4);
  q += __shfl_xor(q, 8);             // all 32 lanes now hold the wave sum
  return q;
}

// Block-wide sum; result valid in wave 0 (in particular threadIdx.x == 0).
__device__ __forceinline__ float block_reduce_sum(float x) {
  __shared__ float smem[32];
  float s = wave_reduce_sum_wmma(x);
  const int lane = threadIdx.x & 31;
  const int wave = threadIdx.x >> 5;
  if (lane == 0) smem[wave] = s;
  __syncthreads();
  float t = 0.0f;
  if (wave == 0) {
    const int nwaves = (blockDim.x + 31) >> 5;
    t = (lane < nwaves) ? smem[lane] : 0.0f;
    t += __shfl_xor(t, 1);
    t += __shfl_xor(t, 2);
    t += __shfl_xor(t, 4);
  }
  return t;
}

// Stage 1: masked reduction of w[i]*log(prob[i*21]) where label[i]==0.
// 4 elements per thread, vectorized label/weight loads; the strided prob
// load is issued only for label==0 lanes (EXEC-masked -> ~1/21 of traffic).
__global__ void pcl_bg_partial(const float* __restrict__ prob,
                               const int*   __restrict__ labels,
                               const float* __restrict__ w,
                               float*       __restrict__ partial,
                               int n) {
  const int i0 = (blockIdx.x * blockDim.x + threadIdx.x) * 4;
  float acc = 0.0f;
  if (i0 + 3 < n) {
    const int4   lab = *(const int4*)(labels + i0);
    const float4 wt  = *(const float4*)(w + i0);
    if (lab.x == 0) acc += wt.x * logf(prob[(size_t)(i0 + 0) * CCH]);
    if (lab.y == 0) acc += wt.y * logf(prob[(size_t)(i0 + 1) * CCH]);
    if (lab.z == 0) acc += wt.z * logf(prob[(size_t)(i0 + 2) * CCH]);
    if (lab.w == 0) acc += wt.w * logf(prob[(size_t)(i0 + 3) * CCH]);
  } else {
    for (int k = 0; k < 4; ++k) {
      const int i = i0 + k;
      if (i < n && labels[i] == 0) acc += w[i] * logf(prob[(size_t)i * CCH]);
    }
  }
  const float bsum = block_reduce_sum(acc);
  if (threadIdx.x == 0) partial[blockIdx.x] = bsum;
}

// Stage 2: reduce block partials, apply gating and the (R==1) fg term.
__global__ void pcl_finalize(const float* __restrict__ partial, int nPartial,
                             const float* __restrict__ pc_probs,
                             const float* __restrict__ img_w,
                             const int*   __restrict__ im_real,
                             float*       __restrict__ out,
                             float invN) {
  float acc = 0.0f;
  for (int i = threadIdx.x; i < nPartial; i += blockDim.x) acc += partial[i];
  float bg = block_reduce_sum(acc);
  if (threadIdx.x == 0) {
    if (im_real[0] == 0) bg = 0.0f;  // gate bg term on im_labels_real[0,0]
    // Reference quirk: R = pc_labels.shape[0] == 1, so only element 0 of
    // pc_probs / img_cls_loss_weights participates in the fg term.
    const float pp0 = pc_probs[0];
    const float w0  = img_w[0];
    bool match = false;
    for (int c = 1; c < CCH; ++c)
      if (im_real[c] != 0 && pp0 == (float)c) match = true;
    const float fg = match ? w0 * logf(pp0) : 0.0f;
    out[0] = (-bg - fg) * invN;
  }
}

extern "C" void kernel_launch(void* const* d_in, const int* in_sizes, int n_in,
                              void* d_out, int out_size, void* d_ws, size_t ws_size,
                              hipStream_t stream) {
  // setup_inputs() order:
  // 0: pcl_prob (N*C f32)      1: labels (N i32)        2: cls_loss_weights (N f32)
  // 3: gt_assignment (unused)  4: pc_labels (unused)    5: pc_probs (P f32)
  // 6: pc_count (unused)       7: img_cls_loss_weights  8: im_labels_real (C i32)
  const float* prob   = (const float*)d_in[0];
  const int*   labels = (const int*)d_in[1];
  const float* clw    = (const float*)d_in[2];
  const float* pcp    = (const float*)d_in[5];
  const float* imgw   = (const float*)d_in[7];
  const int*   imreal = (const int*)d_in[8];

  const int n = in_sizes[1];            // N (labels flat count)
  const int B = 256;                    // 8 waves per block (wave32)
  const int G = (n + B * 4 - 1) / (B * 4);  // 1024 for N = 1M

  float* partial = (float*)d_ws;        // G floats of scratch

  pcl_bg_partial<<<G, B, 0, stream>>>(prob, labels, clw, partial, n);
  pcl_finalize<<<1, B, 0, stream>>>(partial, G, pcp, imgw, imreal,
                                    (float*)d_out, 1.0f / (float)n);
}